// AttPool_20332375179280
// MI455X (gfx1250) — compile-verified
//
#include <hip/hip_runtime.h>
#include <math.h>

// Problem constants
#define DD   768
#define LL   256
#define NMB  256          // N*M = 8*32 blocks
#define HH   3072         // MLP hidden
#define SCALE 0.03608439182435161f  // 1/sqrt(768)

typedef float v2f __attribute__((ext_vector_type(2)));
typedef float v8f __attribute__((ext_vector_type(8)));

// ---------------- wave32 / block reductions ----------------
__device__ __forceinline__ float wave_sum(float v) {
#pragma unroll
  for (int s = 16; s > 0; s >>= 1) v += __shfl_xor(v, s, 32);
  return v;
}
__device__ __forceinline__ float wave_max(float v) {
#pragma unroll
  for (int s = 16; s > 0; s >>= 1) v = fmaxf(v, __shfl_xor(v, s, 32));
  return v;
}
// blockDim.x must be 256 (8 waves)
__device__ __forceinline__ float block_sum(float v, float* sred) {
  const int lane = threadIdx.x & 31, wave = threadIdx.x >> 5;
  v = wave_sum(v);
  if (lane == 0) sred[wave] = v;
  __syncthreads();
  float tot = 0.f;
#pragma unroll
  for (int i = 0; i < 8; ++i) tot += sred[i];
  __syncthreads();
  return tot;
}
__device__ __forceinline__ float block_max(float v, float* sred) {
  const int lane = threadIdx.x & 31, wave = threadIdx.x >> 5;
  v = wave_max(v);
  if (lane == 0) sred[wave] = v;
  __syncthreads();
  float m = -3.4e38f;
#pragma unroll
  for (int i = 0; i < 8; ++i) m = fmaxf(m, sred[i]);
  __syncthreads();
  return m;
}

// ---------------- tiny projections: q = probe*Wq^T + bq ; qk = Wk^T q ----------------
__global__ void proj_q_kernel(const float* __restrict__ probe, const float* __restrict__ wq,
                              const float* __restrict__ bq, float* __restrict__ q) {
  const int d = blockIdx.x * blockDim.x + threadIdx.x;
  if (d >= DD) return;
  const float* wr = wq + d * DD;
  float acc = bq[d];
  for (int e = 0; e < DD; ++e) acc += probe[e] * wr[e];
  q[d] = acc;
}

__global__ void proj_qk_kernel(const float* __restrict__ q, const float* __restrict__ wk,
                               float* __restrict__ qk) {
  const int e = blockIdx.x * blockDim.x + threadIdx.x;
  if (e >= DD) return;
  float acc = 0.f;
  for (int d = 0; d < DD; ++d) acc += q[d] * wk[d * DD + e];  // coalesced across threads
  qk[e] = acc;
}

// ---------------- fused attention pool: one block per (n,m) ----------------
// pass1: logits[l] = (x_l . qk) * SCALE   (wave-per-row, float4 coalesced)
// softmax over L=256 in LDS
// pass2: xbar[d] = sum_l p_l * x[l][d]    (re-read hits L2: 768KB/block working set)
__global__ void attpool_kernel(const float* __restrict__ x, const float* __restrict__ qk,
                               float* __restrict__ xbar) {
  __shared__ float s_qk[DD];
  __shared__ float s_p[LL];
  __shared__ float s_red[8];
  const int t = threadIdx.x;          // 0..255
  const int blk = blockIdx.x;         // 0..255
  const float* xb = x + (size_t)blk * LL * DD;

  for (int i = t; i < DD; i += 256) s_qk[i] = qk[i];
  __syncthreads();

  const int wave = t >> 5, lane = t & 31;
  // pass 1: wave w handles rows [w*32, w*32+32)
  for (int r = 0; r < 32; ++r) {
    const int l = wave * 32 + r;
    const float* xr = xb + l * DD;
    float dot = 0.f;
#pragma unroll
    for (int j = 0; j < 6; ++j) {
      const int off = lane * 4 + j * 128;
      const float4 xv = *(const float4*)(xr + off);
      dot += xv.x * s_qk[off] + xv.y * s_qk[off + 1] +
             xv.z * s_qk[off + 2] + xv.w * s_qk[off + 3];
    }
    dot = wave_sum(dot);
    if (lane == 0) s_p[l] = dot * SCALE;
  }
  __syncthreads();

  // softmax over 256 logits
  const float v = s_p[t];
  const float mx = block_max(v, s_red);
  const float e = __expf(v - mx);
  const float tot = block_sum(e, s_red);
  s_p[t] = e / tot;
  __syncthreads();

  // pass 2: thread t accumulates d = t, t+256, t+512 (coalesced)
  float a0 = 0.f, a1 = 0.f, a2 = 0.f;
  for (int l = 0; l < LL; ++l) {
    const float pl = s_p[l];
    const float* row = xb + l * DD;
    a0 += pl * row[t];
    a1 += pl * row[t + 256];
    a2 += pl * row[t + 512];
  }
  float* ob = xbar + blk * DD;
  ob[t] = a0; ob[t + 256] = a1; ob[t + 512] = a2;
}

// ---------------- fp32 WMMA GEMM: C[256 x ncols] = A[256 x K] * W[ncols x K]^T (+bias/gelu/resid)
// one wave per 16x16 output tile; V_WMMA_F32_16X16X4_F32, K-step = 4.
// A layout: lane m=lane&15 holds K = 2*(lane>>4)+{0,1}; B mirrored with n=lane&15.
// C/D: 8 VGPRs, row m = r + 8*(lane>>4), col n = lane&15.
template <int KDIM, bool GELU_ACT, bool RESID>
__global__ void gemm_wmma_kernel(const float* __restrict__ A, const float* __restrict__ W,
                                 const float* __restrict__ bias, const float* __restrict__ resid,
                                 float* __restrict__ C, int ncols) {
  const int lane = threadIdx.x & 31;
  const int wave = threadIdx.x >> 5;
  const int tile = blockIdx.x * 4 + wave;
  const int tilesPerRow = ncols >> 4;
  const int rt = tile / tilesPerRow;
  const int ct = tile - rt * tilesPerRow;

  const int mn = lane & 15;
  const int khalf = (lane >> 4) << 1;  // 0 or 2
  const float* arow = A + (size_t)(rt * 16 + mn) * KDIM + khalf;
  const float* brow = W + (size_t)(ct * 16 + mn) * KDIM + khalf;

  v8f c = {0.f, 0.f, 0.f, 0.f, 0.f, 0.f, 0.f, 0.f};
#pragma unroll 8
  for (int k = 0; k < KDIM; k += 4) {
    const v2f a = *(const v2f*)(arow + k);
    const v2f b = *(const v2f*)(brow + k);
    c = __builtin_amdgcn_wmma_f32_16x16x4_f32(false, a, false, b, (short)0, c, false, false);
  }

  const int ng = ct * 16 + mn;
  const int mbase = rt * 16 + ((lane >> 4) << 3);
  const float bn = bias[ng];
#pragma unroll
  for (int r = 0; r < 8; ++r) {
    const int mg = mbase + r;
    float val = c[r] + bn;
    if (GELU_ACT) val = 0.5f * val * (1.f + erff(val * 0.70710678118654752f));  // exact gelu
    if (RESID) val += resid[(size_t)mg * ncols + ng];
    C[(size_t)mg * ncols + ng] = val;
  }
}

// ---------------- LayerNorm over D=768, one block (256 thr) per row ----------------
__global__ void layernorm_kernel(const float* __restrict__ xo, const float* __restrict__ gamma,
                                 const float* __restrict__ beta, float* __restrict__ y) {
  __shared__ float s_red[8];
  const int row = blockIdx.x, t = threadIdx.x;
  const float* xr = xo + (size_t)row * DD;
  const float a0 = xr[t], a1 = xr[t + 256], a2 = xr[t + 512];
  const float mu = block_sum(a0 + a1 + a2, s_red) * (1.f / 768.f);
  const float d0 = a0 - mu, d1 = a1 - mu, d2 = a2 - mu;
  const float var = block_sum(d0 * d0 + d1 * d1 + d2 * d2, s_red) * (1.f / 768.f);
  const float inv = rsqrtf(var + 1e-5f);
  float* yr = y + (size_t)row * DD;
  yr[t]       = d0 * inv * gamma[t]       + beta[t];
  yr[t + 256] = d1 * inv * gamma[t + 256] + beta[t + 256];
  yr[t + 512] = d2 * inv * gamma[t + 512] + beta[t + 512];
}

extern "C" void kernel_launch(void* const* d_in, const int* in_sizes, int n_in,
                              void* d_out, int out_size, void* d_ws, size_t ws_size,
                              hipStream_t stream) {
  (void)in_sizes; (void)n_in; (void)out_size; (void)ws_size;
  const float* x     = (const float*)d_in[0];
  const float* probe = (const float*)d_in[1];
  const float* wq    = (const float*)d_in[2];
  const float* bq    = (const float*)d_in[3];
  const float* wk    = (const float*)d_in[4];
  // d_in[5] = bk: adds q.bk to every logit -> softmax-invariant, unused
  const float* wv    = (const float*)d_in[6];
  const float* bv    = (const float*)d_in[7];
  const float* gamma = (const float*)d_in[8];
  const float* beta  = (const float*)d_in[9];
  const float* w1    = (const float*)d_in[10];
  const float* b1    = (const float*)d_in[11];
  const float* w2    = (const float*)d_in[12];
  const float* b2    = (const float*)d_in[13];
  float* out = (float*)d_out;

  float* ws   = (float*)d_ws;
  float* q    = ws;                 // 768
  float* qk   = q + DD;             // 768
  float* xbar = qk + DD;            // 256*768
  float* xo   = xbar + NMB * DD;    // 256*768
  float* y    = xo + NMB * DD;      // 256*768
  float* h    = y + NMB * DD;       // 256*3072   (total ~5.5 MB)

  proj_q_kernel <<<3, 256, 0, stream>>>(probe, wq, bq, q);
  proj_qk_kernel<<<3, 256, 0, stream>>>(q, wk, qk);
  attpool_kernel<<<NMB, 256, 0, stream>>>(x, qk, xbar);

  // xo = xbar @ wv^T + bv          (256 x 768, K=768)
  gemm_wmma_kernel<DD, false, false><<<(16 * (DD / 16)) / 4, 128, 0, stream>>>(
      xbar, wv, bv, nullptr, xo, DD);
  // y = LN(xo)
  layernorm_kernel<<<NMB, 256, 0, stream>>>(xo, gamma, beta, y);
  // h = gelu(y @ w1^T + b1)        (256 x 3072, K=768)
  gemm_wmma_kernel<DD, true, false><<<(16 * (HH / 16)) / 4, 128, 0, stream>>>(
      y, w1, b1, nullptr, h, HH);
  // out = xo + h @ w2^T + b2       (256 x 768, K=3072)
  gemm_wmma_kernel<HH, false, true><<<(16 * (DD / 16)) / 4, 128, 0, stream>>>(
      h, w2, b2, xo, out, DD);
}